// DenseGTVConv_28243704938878
// MI455X (gfx1250) — compile-verified
//
#include <hip/hip_runtime.h>

// Problem constants (from reference setup_inputs)
#define B_    4
#define N_    1024
#define FI_   128
#define FO_   64
#define DELTA_ 1.0f
#define EPS_   1e-3f

typedef __attribute__((ext_vector_type(2))) float v2f;
typedef __attribute__((ext_vector_type(8))) float v8f;

// D(16x16,f32) = A(16x4,f32) x B(4x16,f32) + C  — native CDNA5 fp32 WMMA
static __device__ __forceinline__ v8f wmma_f32_16x16x4(v2f a, v2f b, v8f c) {
  return __builtin_amdgcn_wmma_f32_16x16x4_f32(
      /*neg_a=*/false, a, /*neg_b=*/false, b,
      /*c_mod=*/(short)0, c, /*reuse_a=*/false, /*reuse_b=*/false);
}

// ---------------------------------------------------------------------------
// Kernel 1: xw[4096,64] = x[4096,128] @ W[128,64]   (fp32 WMMA, K in steps of 4)
// grid = 256 blocks (16 rows each), block = 128 threads (4 waves, one 16-col
// f-chunk per wave).
// A layout: lane<16 -> row=lane, k=ks..ks+1 ; lane>=16 -> row=lane-16, k=ks+2..
// B layout mirrors A with K as the "row" dim; C/D: VGPR r -> row r (+8 for hi half).
// ---------------------------------------------------------------------------
__global__ void __launch_bounds__(128)
gtv_xw_gemm(const float* __restrict__ x, const float* __restrict__ w,
            float* __restrict__ xw) {
  const int tid  = threadIdx.x;
  const int wv   = tid >> 5;
  const int lane = tid & 31;
  const int hh   = lane >> 4;   // half-wave select
  const int l15  = lane & 15;
  const int rt   = blockIdx.x;  // 16-row tile over flattened B*N
  const int f0   = wv * 16;

  const float* __restrict__ xrow = x + (size_t)(rt * 16 + l15) * FI_;
  v8f c = {0.f, 0.f, 0.f, 0.f, 0.f, 0.f, 0.f, 0.f};

  #pragma unroll 4
  for (int kc = 0; kc < FI_ / 4; ++kc) {
    const int ks = kc * 4 + hh * 2;
    v2f a;  a.x = xrow[ks];  a.y = xrow[ks + 1];
    const float* __restrict__ wp = w + (size_t)ks * FO_ + f0 + l15;
    v2f b;  b.x = wp[0];     b.y = wp[FO_];
    c = wmma_f32_16x16x4(a, b, c);
  }

  float* __restrict__ orow = xw + (size_t)(rt * 16 + hh * 8) * FO_ + f0 + l15;
  #pragma unroll
  for (int r = 0; r < 8; ++r) orow[(size_t)r * FO_] = c[r];
}

// ---------------------------------------------------------------------------
// Kernel 2 (fused): per (b, i-tile of 16 rows):
//   loop j-tiles: L1-distance tile (VALU, C-layout) -> mod_adj -> deg partial
//   -> WMMA accumulate out[16x64] += (DELTA*mod_adj)[16x16] @ xw_j[16x64]
// then cross-wave LDS reduction, diagonal term + bias on store.
// grid = B*(N/16) = 256 blocks, block = 256 threads (8 waves, 8 j-tiles each).
// ---------------------------------------------------------------------------
__global__ void __launch_bounds__(256)
gtv_fused(const float* __restrict__ adj, const float* __restrict__ xw,
          const float* __restrict__ bias, float* __restrict__ out) {
  __shared__ float xwi_s[16][FO_];      // this block's xw_i tile
  __shared__ float Msh[8][16][20];      // per-wave mod_adj tile (pad 20: no bank conflicts)
  __shared__ float out_s[16][FO_];      // cross-wave output reduction
  __shared__ float deg_s[16];           // cross-wave degree reduction

  const int tid  = threadIdx.x;
  const int wv   = tid >> 5;
  const int lane = tid & 31;
  const int hh   = lane >> 4;
  const int l15  = lane & 15;

  const int b  = blockIdx.x >> 6;       // 64 i-tiles per batch
  const int i0 = (blockIdx.x & 63) * 16;

  // init shared: zero out_s/deg_s, stage xw_i tile (4 floats per thread)
  if (tid < 16) deg_s[tid] = 0.f;
  {
    const int row = tid >> 4;
    const int c4  = (tid & 15) * 4;
    *(float4*)&out_s[row][c4] = make_float4(0.f, 0.f, 0.f, 0.f);
    *(float4*)&xwi_s[row][c4] =
        *(const float4*)(xw + (size_t)(b * N_ + i0 + row) * FO_ + c4);
  }
  __syncthreads();

  float dacc[8];
  #pragma unroll
  for (int r = 0; r < 8; ++r) dacc[r] = 0.f;
  v8f oacc[4];
  #pragma unroll
  for (int f = 0; f < 4; ++f)
    #pragma unroll
    for (int e = 0; e < 8; ++e) oacc[f][e] = 0.f;

  const float* __restrict__ adjb = adj + (size_t)b * N_ * N_ + (size_t)i0 * N_;
  const float* __restrict__ xwb  = xw + (size_t)b * N_ * FO_;

  for (int step = 0; step < 8; ++step) {
    const int jt = step * 8 + wv;
    const int j0 = jt * 16;
    if (step < 7)  // hide latency of next adj tile
      __builtin_prefetch(adjb + (size_t)(hh * 8) * N_ + (j0 + 128) + l15, 0, 0);

    // ---- pairwise L1 distances, directly in WMMA C-layout ----
    float acc[8];
    #pragma unroll
    for (int r = 0; r < 8; ++r) acc[r] = 0.f;
    const float* __restrict__ xwj = xwb + (size_t)(j0 + l15) * FO_;
    #pragma unroll 4
    for (int kc = 0; kc < FO_ / 4; ++kc) {
      const float4 xj = *(const float4*)(xwj + kc * 4);
      #pragma unroll
      for (int r = 0; r < 8; ++r) {
        const float4 xi = *(const float4*)&xwi_s[r + hh * 8][kc * 4];
        acc[r] += fabsf(xi.x - xj.x) + fabsf(xi.y - xj.y)
                + fabsf(xi.z - xj.z) + fabsf(xi.w - xj.w);
      }
    }

    // ---- modified adjacency tile (DELTA folded in), deg partial ----
    const float* __restrict__ ap = adjb + j0 + l15;
    #pragma unroll
    for (int r = 0; r < 8; ++r) {
      const int irow = r + hh * 8;
      const float a = ap[(size_t)irow * N_];
      float m = (a != 0.f) ? (a / fmaxf(acc[r], EPS_)) : a;
      m *= DELTA_;
      dacc[r] += m;
      Msh[wv][irow][l15] = m;  // bounce to LDS to re-read in A-matrix layout
    }

    // ---- WMMA: oacc[16x64] += Mtile[16x16] @ xw_j[16x64] (fp32, K=4 steps) ----
    v2f avec[4];
    #pragma unroll
    for (int kchunk = 0; kchunk < 4; ++kchunk) {
      const int ks = kchunk * 4 + hh * 2;
      avec[kchunk].x = Msh[wv][l15][ks];
      avec[kchunk].y = Msh[wv][l15][ks + 1];
    }
    #pragma unroll
    for (int fc = 0; fc < 4; ++fc) {
      #pragma unroll
      for (int kchunk = 0; kchunk < 4; ++kchunk) {
        const int ks = kchunk * 4 + hh * 2;
        const float* __restrict__ bp =
            xwb + (size_t)(j0 + ks) * FO_ + fc * 16 + l15;  // L2-resident
        v2f bv;  bv.x = bp[0];  bv.y = bp[FO_];
        oacc[fc] = wmma_f32_16x16x4(avec[kchunk], bv, oacc[fc]);
      }
    }
  }

  // ---- cross-wave reductions (LDS float atomics -> ds_add_f32) ----
  #pragma unroll
  for (int r = 0; r < 8; ++r) {
    float d = dacc[r];
    d += __shfl_xor(d, 1, 16);
    d += __shfl_xor(d, 2, 16);
    d += __shfl_xor(d, 4, 16);
    d += __shfl_xor(d, 8, 16);
    if (l15 == 0) atomicAdd(&deg_s[r + hh * 8], d);
  }
  #pragma unroll
  for (int fc = 0; fc < 4; ++fc)
    #pragma unroll
    for (int r = 0; r < 8; ++r)
      atomicAdd(&out_s[r + hh * 8][fc * 16 + l15], oacc[fc][r]);
  __syncthreads();

  // ---- finalize: out = DELTA*mod_adj@xw + (1 - DELTA*deg)*xw_i + bias ----
  {
    const int row = tid >> 4;
    const int c4  = (tid & 15) * 4;
    const float diagc = 1.f - deg_s[row];   // deg_s already has DELTA folded
    const float4 xi = *(const float4*)&xwi_s[row][c4];
    const float4 bs = *(const float4*)(bias + c4);
    float4 o = *(const float4*)&out_s[row][c4];
    o.x += diagc * xi.x + bs.x;
    o.y += diagc * xi.y + bs.y;
    o.z += diagc * xi.z + bs.z;
    o.w += diagc * xi.w + bs.w;
    *(float4*)(out + (size_t)(b * N_ + i0 + row) * FO_ + c4) = o;
  }
}

extern "C" void kernel_launch(void* const* d_in, const int* in_sizes, int n_in,
                              void* d_out, int out_size, void* d_ws, size_t ws_size,
                              hipStream_t stream) {
  (void)in_sizes; (void)n_in; (void)out_size; (void)ws_size;
  const float* x    = (const float*)d_in[0];   // [4,1024,128]
  const float* adj  = (const float*)d_in[1];   // [4,1024,1024]
  const float* w    = (const float*)d_in[2];   // [128,64]
  const float* bias = (const float*)d_in[3];   // [64]
  float*       out  = (float*)d_out;           // [4,1024,64]
  float*       xw   = (float*)d_ws;            // scratch: 4096*64 f32 = 1 MB

  gtv_xw_gemm<<<(B_ * N_) / 16, 128, 0, stream>>>(x, w, xw);
  gtv_fused  <<<B_ * (N_ / 16), 256, 0, stream>>>(adj, xw, bias, out);
}